// GraphConvNet_25649544692460
// MI455X (gfx1250) — compile-verified
//
#include <hip/hip_runtime.h>
#include <math.h>

#define N_NODES  100000
#define N_EDGES  3200000
#define N_GRAPHS 1000
#define C  11
#define D  4
#define ZD 26      // 2C + D
#define ZP 28      // K padded to multiple of 4
#define NSTEPS 7   // ZP / 4
#define BN_EPS 1e-5f

typedef __attribute__((ext_vector_type(2))) float v2f;
typedef __attribute__((ext_vector_type(8))) float v8f;

// Fast gate math on the CDNA5 TRANS pipe (v_exp_f32 / v_log_f32 / v_rcp_f32):
// co-executes with VALU/WMMA, ~10x fewer instructions than libm expf/log1pf.
__device__ __forceinline__ float sigmoidf_(float x) {
    return __builtin_amdgcn_rcpf(1.0f + __expf(-x));
}
__device__ __forceinline__ float softplusf_(float x) {
    return (x > 15.0f) ? x : __logf(1.0f + __expf(x));
}

// -------------------------------------------------------------------------
// Edge-message kernel: per wave, a 16-edge tile.
//   Z (16 x 28, f32, LDS)  @  W (28 x 16, f32, LDS)  via V_WMMA_F32_16X16X4_F32
//   accumulators seeded with bias; two parallel accumulators (lin_f, lin_s).
//   msg = sigmoid(F) * softplus(S); atomicAdd scatter into agg[dst].
// -------------------------------------------------------------------------
__global__ __launch_bounds__(256)
void cgconv_edge_wmma(const float* __restrict__ X,
                      const int*   __restrict__ srcIdx,
                      const int*   __restrict__ dstIdx,
                      const float* __restrict__ EA,
                      const float* __restrict__ Wf, const float* __restrict__ bf,
                      const float* __restrict__ Ws, const float* __restrict__ bs,
                      float* __restrict__ agg)
{
    __shared__ float shZ[8][16][ZP];   // 8 waves x 16 edges x 28 features
    __shared__ float shW[2][ZP][16];   // padded Wf / Ws
    __shared__ int   shD[8][16];       // dst node per edge-in-tile

    const int tid  = threadIdx.x;
    const int wave = tid >> 5;
    const int lane = tid & 31;
    const int m    = lane & 15;   // row (edge in tile) / output col
    const int half = lane >> 4;   // K-pair selector per ISA A/B layout

    // Stage padded weight matrices once per block (26x11 -> 28x16, zero pad)
    for (int idx = tid; idx < 2 * ZP * 16; idx += 256) {
        int mat = idx / (ZP * 16);
        int rem = idx % (ZP * 16);
        int r = rem >> 4;
        int n = rem & 15;
        const float* Wsrc = mat ? Ws : Wf;
        shW[mat][r][n] = (r < ZD && n < C) ? Wsrc[r * C + n] : 0.0f;
    }

    // Stage this wave's 16-edge Z tile: z = [x_dst | x_src | e], zero padded
    const int tile = blockIdx.x * 8 + wave;
    const int e    = tile * 16 + m;
    const bool ve  = (e < N_EDGES);
    const int ec   = ve ? e : (N_EDGES - 1);
    const int sN   = srcIdx[ec];
    const int dN   = dstIdx[ec];
    if (half == 0) shD[wave][m] = dN;
    #pragma unroll
    for (int k = 0; k < 14; ++k) {
        int kk = half * 14 + k;          // half 0 -> 0..13, half 1 -> 14..27
        float v;
        if (kk < C)          v = X[dN * C + kk];
        else if (kk < 2 * C) v = X[sN * C + (kk - C)];
        else if (kk < ZD)    v = EA[ec * D + (kk - 2 * C)];
        else                 v = 0.0f;
        shZ[wave][m][kk] = ve ? v : 0.0f;
    }
    __syncthreads();

    // Seed accumulators with bias broadcast along rows (cols >= C are 0)
    const int n = m;                      // output column for this lane
    const float bfv = (n < C) ? bf[n] : 0.0f;
    const float bsv = (n < C) ? bs[n] : 0.0f;
    v8f accF, accS;
    #pragma unroll
    for (int i = 0; i < 8; ++i) { accF[i] = bfv; accS[i] = bsv; }

    // 7 K-steps of f32 WMMA for each of the two linear maps
    #pragma unroll
    for (int st = 0; st < NSTEPS; ++st) {
        const int ka = st * 4 + half * 2;
        v2f a;  a.x  = shZ[wave][m][ka]; a.y  = shZ[wave][m][ka + 1];
        v2f bF; bF.x = shW[0][ka][n];    bF.y = shW[0][ka + 1][n];
        v2f bS; bS.x = shW[1][ka][n];    bS.y = shW[1][ka + 1][n];
        accF = __builtin_amdgcn_wmma_f32_16x16x4_f32(
                   false, a, false, bF, (short)0, accF, false, false);
        accS = __builtin_amdgcn_wmma_f32_16x16x4_f32(
                   false, a, false, bS, (short)0, accS, false, false);
    }

    // Gate + scatter-add (L2-resident atomics)
    if (n < C) {
        #pragma unroll
        for (int v = 0; v < 8; ++v) {
            int row = v + half * 8;
            int ge  = tile * 16 + row;
            if (ge < N_EDGES) {
                float msg = sigmoidf_(accF[v]) * softplusf_(accS[v]);
                atomicAdd(&agg[shD[wave][row] * C + n], msg);
            }
        }
    }
}

// -------------------------------------------------------------------------
// Per-channel sum / sumsq over nodes (BN statistics)
// -------------------------------------------------------------------------
__global__ __launch_bounds__(256)
void bn_stats(const float* __restrict__ agg, float* __restrict__ stats)
{
    __shared__ float sh[2 * C];
    int tid = threadIdx.x;
    if (tid < 2 * C) sh[tid] = 0.0f;
    __syncthreads();

    float s[C], q[C];
    #pragma unroll
    for (int c = 0; c < C; ++c) { s[c] = 0.0f; q[c] = 0.0f; }

    for (int node = blockIdx.x * blockDim.x + tid; node < N_NODES;
         node += gridDim.x * blockDim.x) {
        #pragma unroll
        for (int c = 0; c < C; ++c) {
            float v = agg[node * C + c];
            s[c] += v;
            q[c] += v * v;
        }
    }
    #pragma unroll
    for (int c = 0; c < C; ++c) {
        atomicAdd(&sh[c],     s[c]);
        atomicAdd(&sh[C + c], q[c]);
    }
    __syncthreads();
    if (tid < 2 * C) atomicAdd(&stats[tid], sh[tid]);
}

// -------------------------------------------------------------------------
// BatchNorm apply + residual: h = x + (agg - mu) * rsqrt(var+eps)*gamma + beta
// -------------------------------------------------------------------------
__global__ __launch_bounds__(256)
void bn_apply(const float* __restrict__ xin, const float* __restrict__ agg,
              const float* __restrict__ stats,
              const float* __restrict__ gamma, const float* __restrict__ beta,
              float* __restrict__ hout)
{
    int idx = blockIdx.x * blockDim.x + threadIdx.x;
    if (idx >= N_NODES * C) return;
    int c = idx % C;
    float mu  = stats[c] * (1.0f / N_NODES);
    float var = stats[C + c] * (1.0f / N_NODES) - mu * mu;
    float nrm = (agg[idx] - mu) * rsqrtf(var + BN_EPS) * gamma[c] + beta[c];
    hout[idx] = xin[idx] + nrm;
}

// -------------------------------------------------------------------------
// Per-graph mean-pool accumulation
// -------------------------------------------------------------------------
__global__ __launch_bounds__(256)
void pool_kernel(const float* __restrict__ h, const int* __restrict__ batch,
                 float* __restrict__ gsum, float* __restrict__ gcnt)
{
    int node = blockIdx.x * blockDim.x + threadIdx.x;
    if (node >= N_NODES) return;
    int g = batch[node];
    #pragma unroll
    for (int c = 0; c < C; ++c)
        atomicAdd(&gsum[g * C + c], h[node * C + c]);
    atomicAdd(&gcnt[g], 1.0f);
}

// -------------------------------------------------------------------------
// MLP head: softplus(pooled @ W1 + b1) @ W2 + b2   (one thread per graph)
// -------------------------------------------------------------------------
__global__ __launch_bounds__(256)
void head_kernel(const float* __restrict__ gsum, const float* __restrict__ gcnt,
                 const float* __restrict__ W1, const float* __restrict__ b1,
                 const float* __restrict__ W2, const float* __restrict__ b2,
                 float* __restrict__ out)
{
    int g = blockIdx.x * blockDim.x + threadIdx.x;
    if (g >= N_GRAPHS) return;
    float inv = 1.0f / fmaxf(gcnt[g], 1.0f);
    float p[C];
    #pragma unroll
    for (int c = 0; c < C; ++c) p[c] = gsum[g * C + c] * inv;
    float acc = b2[0];
    #pragma unroll
    for (int j = 0; j < 5; ++j) {
        float hsum = b1[j];
        #pragma unroll
        for (int c = 0; c < C; ++c) hsum += p[c] * W1[c * 5 + j];
        acc += softplusf_(hsum) * W2[j];
    }
    out[g] = acc;
}

__global__ __launch_bounds__(256)
void zero_kernel(float* __restrict__ p, int count)
{
    int i = blockIdx.x * blockDim.x + threadIdx.x;
    if (i < count) p[i] = 0.0f;
}

// -------------------------------------------------------------------------
extern "C" void kernel_launch(void* const* d_in, const int* in_sizes, int n_in,
                              void* d_out, int out_size, void* d_ws, size_t ws_size,
                              hipStream_t stream)
{
    const float* x     = (const float*)d_in[0];
    const int*   ei    = (const int*)  d_in[1];
    const float* ea    = (const float*)d_in[2];
    const int*   batch = (const int*)  d_in[3];
    const float *Wf1 = (const float*)d_in[4],  *bf1 = (const float*)d_in[5];
    const float *Ws1 = (const float*)d_in[6],  *bs1 = (const float*)d_in[7];
    const float *g1  = (const float*)d_in[8],  *be1 = (const float*)d_in[9];
    const float *Wf2 = (const float*)d_in[10], *bf2 = (const float*)d_in[11];
    const float *Ws2 = (const float*)d_in[12], *bs2 = (const float*)d_in[13];
    const float *g2  = (const float*)d_in[14], *be2 = (const float*)d_in[15];
    const float *Wc1 = (const float*)d_in[16], *bc1 = (const float*)d_in[17];
    const float *Wc2 = (const float*)d_in[18], *bc2 = (const float*)d_in[19];
    float* out = (float*)d_out;

    const int* srcIdx = ei;             // edge_index[0]
    const int* dstIdx = ei + N_EDGES;   // edge_index[1]

    float* ws    = (float*)d_ws;
    float* agg   = ws;                               // N*C
    float* h1    = agg   + (size_t)N_NODES * C;      // N*C
    float* h2    = h1    + (size_t)N_NODES * C;      // N*C
    float* stats = h2    + (size_t)N_NODES * C;      // 2*C
    float* gsum  = stats + 2 * C;                    // G*C (gcnt follows)
    float* gcnt  = gsum  + (size_t)N_GRAPHS * C;     // G

    const int NC = N_NODES * C;
    const dim3 b256(256);
    const int edgeBlocks = (N_EDGES / 16 + 7) / 8;   // 8 tiles (waves) per block

    // ---- layer 1 ----
    zero_kernel<<<(NC + 255) / 256, b256, 0, stream>>>(agg, NC);
    zero_kernel<<<1, 32, 0, stream>>>(stats, 2 * C);
    cgconv_edge_wmma<<<edgeBlocks, b256, 0, stream>>>(
        x, srcIdx, dstIdx, ea, Wf1, bf1, Ws1, bs1, agg);
    bn_stats<<<256, b256, 0, stream>>>(agg, stats);
    bn_apply<<<(NC + 255) / 256, b256, 0, stream>>>(x, agg, stats, g1, be1, h1);

    // ---- layer 2 ----
    zero_kernel<<<(NC + 255) / 256, b256, 0, stream>>>(agg, NC);
    zero_kernel<<<1, 32, 0, stream>>>(stats, 2 * C);
    cgconv_edge_wmma<<<edgeBlocks, b256, 0, stream>>>(
        h1, srcIdx, dstIdx, ea, Wf2, bf2, Ws2, bs2, agg);
    bn_stats<<<256, b256, 0, stream>>>(agg, stats);
    bn_apply<<<(NC + 255) / 256, b256, 0, stream>>>(h1, agg, stats, g2, be2, h2);

    // ---- pool + head ----
    zero_kernel<<<(N_GRAPHS * C + N_GRAPHS + 255) / 256, b256, 0, stream>>>(
        gsum, N_GRAPHS * C + N_GRAPHS);
    pool_kernel<<<(N_NODES + 255) / 256, b256, 0, stream>>>(h2, batch, gsum, gcnt);
    head_kernel<<<(N_GRAPHS + 255) / 256, b256, 0, stream>>>(
        gsum, gcnt, Wc1, bc1, Wc2, bc2, out);
}